// DWT_2963527434289
// MI455X (gfx1250) — compile-verified
//
#include <hip/hip_runtime.h>

// ---------------------------------------------------------------------------
// 2D Haar DWT, (8,64,512,512) f32 -> 4x (8,64,256,256) f32.
// Per 16x16 tile X:  Y = Hr * X * Hr^T  via V_WMMA_F32_16X16X4_F32 (8 WMMAs),
// computed as U = X*Bc (Bc = Hr^T), then Z = U^T*Bc = Y^T.
// Input staged global->LDS with global_load_async_to_lds_b128, double buffered.
// Memory bound: ~1.07GB traffic -> ~46us floor at 23.3 TB/s.
// ---------------------------------------------------------------------------

typedef float v2f __attribute__((ext_vector_type(2)));
typedef float v4f __attribute__((ext_vector_type(4)));
typedef float v8f __attribute__((ext_vector_type(8)));
typedef int   v4i __attribute__((ext_vector_type(4)));

#ifndef __has_builtin
#define __has_builtin(x) 0
#endif

#if __has_builtin(__builtin_amdgcn_global_load_async_to_lds_b128)
#define DWT_ASYNC 1
#else
#define DWT_ASYNC 0
#endif

#if DWT_ASYNC
#if __has_builtin(__builtin_amdgcn_s_wait_asynccnt)
#define DWT_WAIT_ASYNC(n)                       \
  do {                                          \
    __builtin_amdgcn_s_wait_asynccnt(n);        \
    asm volatile("" ::: "memory");              \
  } while (0)
#else
#define DWT_WAIT_ASYNC(n) asm volatile("s_wait_asynccnt %0" ::"i"(n) : "memory")
#endif
#else
#define DWT_WAIT_ASYNC(n) asm volatile("" ::: "memory")
#endif

#define IMGS 512            // b*c = 8*64
#define TILES_TOTAL 524288  // 512 imgs * 32 * 32 tiles
#define TPW 8               // tiles per wave
#define LDS_S 20            // padded LDS row stride in floats (bank-conflict-free, 16B aligned)
#define LDS_TILE (16 * LDS_S)
#define BAND_N 33554432ULL  // 512*256*256

typedef __attribute__((address_space(1))) v4i gl_v4i;
typedef __attribute__((address_space(3))) v4i lds_v4i;

__device__ __forceinline__ void stage_tile(const float* __restrict__ x, int T,
                                           float* __restrict__ lbuf, int lane) {
  const int q   = T & 31;
  const int p   = (T >> 5) & 31;
  const int img = T >> 10;
  const float* g0 = x + ((size_t)img << 18) + ((size_t)(p << 4) << 9) + (q << 4);
#pragma unroll
  for (int i = 0; i < 2; ++i) {
    const int slot = i * 32 + lane;
    const int rr = slot >> 2;
    const int cc = slot & 3;
    const float* g = g0 + rr * 512 + cc * 4;
    float* l = lbuf + rr * LDS_S + cc * 4;
#if DWT_ASYNC
    __builtin_amdgcn_global_load_async_to_lds_b128((gl_v4i*)g, (lds_v4i*)l, 0, 0);
#else
    *(v4f*)l = __builtin_nontemporal_load((const v4f*)g);
#endif
  }
}

__global__ __launch_bounds__(256) void dwt_haar_wmma_kernel(
    const float* __restrict__ x, float* __restrict__ out) {
  __shared__ float smem[8 * 2 * LDS_TILE];  // 8 waves * 2 buffers * padded tile

  const int lane = threadIdx.x & 31;
  const int wv   = threadIdx.x >> 5;
  float* wbuf    = &smem[wv * 2 * LDS_TILE];

  // ---- constant B operands: Bc = Hr^T rows 4k..4k+3, in WMMA B-layout -----
  // lane<16 holds rows {4k,4k+1} in VGPR{0,1}; lane>=16 holds rows {4k+2,4k+3}
  const float r = 0.70710678118654752440f;
  const int n16 = lane & 15;
  const int hi2 = (lane >> 4) << 1;
  v2f bk[4];
#pragma unroll
  for (int k = 0; k < 4; ++k) {
#pragma unroll
    for (int v = 0; v < 2; ++v) {
      const int w = 4 * k + hi2 + v;  // row index of Bc
      float val = 0.0f;
      if (n16 == (w >> 1)) val = r;                       // low-pass column
      if (n16 == 8 + (w >> 1)) val = (w & 1) ? -r : r;    // high-pass column
      bk[k][v] = val;
    }
  }

  const int T0 = (blockIdx.x * 8 + wv) * TPW;

  stage_tile(x, T0, wbuf, lane);  // prologue: prefetch tile 0 into buffer 0

  const int arow = lane & 15;
  const int ash  = (lane >> 4) << 1;   // 0 for lanes 0-15, 2 for lanes 16-31
  const bool lo  = lane < 16;

  for (int t = 0; t < TPW; ++t) {
    if (t + 1 < TPW) {
      stage_tile(x, T0 + t + 1, wbuf + ((t + 1) & 1) * LDS_TILE, lane);
      DWT_WAIT_ASYNC(2);  // tile t's 2 async ops retired (in-order completion)
    } else {
      DWT_WAIT_ASYNC(0);
    }

    const float* lbuf = wbuf + (t & 1) * LDS_TILE;
    const int T   = T0 + t;
    const int q   = T & 31;
    const int p   = (T >> 5) & 31;
    const int img = T >> 10;

    // ---- step 1: U = X * Bc. X chunks as A-operand read from LDS ----------
    v8f U = {0.f, 0.f, 0.f, 0.f, 0.f, 0.f, 0.f, 0.f};
#pragma unroll
    for (int k = 0; k < 4; ++k) {
      v2f a = *(const v2f*)(lbuf + arow * LDS_S + 4 * k + ash);  // ds_load_b64
      U = __builtin_amdgcn_wmma_f32_16x16x4_f32(false, a, false, bk[k],
                                                (short)0, U, false, false);
    }

    // ---- step 2: Z = U^T * Bc = Y^T. Build U^T A-chunks via half swaps ----
    float sh[8];
#pragma unroll
    for (int i = 0; i < 8; ++i) sh[i] = __shfl_xor(U[i], 16, 32);

    v8f Z = {0.f, 0.f, 0.f, 0.f, 0.f, 0.f, 0.f, 0.f};
    v2f a2;
    a2.x = lo ? U[0] : sh[2];
    a2.y = lo ? U[1] : sh[3];
    Z = __builtin_amdgcn_wmma_f32_16x16x4_f32(false, a2, false, bk[0], (short)0, Z, false, false);
    a2.x = lo ? U[4] : sh[6];
    a2.y = lo ? U[5] : sh[7];
    Z = __builtin_amdgcn_wmma_f32_16x16x4_f32(false, a2, false, bk[1], (short)0, Z, false, false);
    a2.x = lo ? sh[0] : U[2];
    a2.y = lo ? sh[1] : U[3];
    Z = __builtin_amdgcn_wmma_f32_16x16x4_f32(false, a2, false, bk[2], (short)0, Z, false, false);
    a2.x = lo ? sh[4] : U[6];
    a2.y = lo ? sh[5] : U[7];
    Z = __builtin_amdgcn_wmma_f32_16x16x4_f32(false, a2, false, bk[3], (short)0, Z, false, false);

    // ---- store: lane holds one full 8-float band row of this tile ---------
    // lanes 0-7: LL, 8-15: HL, 16-23: LH, 24-31: HH
    const int row  = lane & 7;
    const int band = (lane >> 4) + (((lane >> 3) & 1) << 1);  // LL=0,LH=1,HL=2,HH=3
    float* ob = out + (size_t)band * BAND_N + ((size_t)img << 16) +
                ((size_t)(p * 8 + row) << 8) + (size_t)(q * 8);
    v4f z0 = {Z[0], Z[1], Z[2], Z[3]};
    v4f z1 = {Z[4], Z[5], Z[6], Z[7]};
    __builtin_nontemporal_store(z0, (v4f*)ob);
    __builtin_nontemporal_store(z1, (v4f*)(ob + 4));
  }
}

extern "C" void kernel_launch(void* const* d_in, const int* in_sizes, int n_in,
                              void* d_out, int out_size, void* d_ws, size_t ws_size,
                              hipStream_t stream) {
  (void)in_sizes; (void)n_in; (void)d_ws; (void)ws_size; (void)out_size;
  const float* x = (const float*)d_in[0];  // Haar matrices d_in[1..4] are fixed constants
  float* out = (float*)d_out;
  // 524288 tiles / (8 tiles/wave * 8 waves/block) = 8192 blocks
  dwt_haar_wmma_kernel<<<dim3(8192), dim3(256), 0, stream>>>(x, out);
}